// GCNNet_22694607192485
// MI455X (gfx1250) — compile-verified
//
#include <hip/hip_runtime.h>
#include <math.h>

// GCN 2-layer forward for MI455X (gfx1250, wave32).
// x:[N,512] @ W1:[512,16] -> scatter-add over edges -> relu -> @ W2:[16,40]
// -> scatter-add -> log_softmax. GEMMs use V_WMMA_F32_16X16X4_F32.

typedef float v2f __attribute__((ext_vector_type(2)));
typedef float v8f __attribute__((ext_vector_type(8)));

#define F_IN 512
#define HID  16
#define NCLS 40

// ---------------- utility kernels ----------------

__global__ void fill_kernel(float* __restrict__ p, float v, long n) {
  long i = (long)blockIdx.x * blockDim.x + threadIdx.x;
  if (i < n) p[i] = v;
}

// deg[dst] += ew  (deg pre-initialized to 1.0 == self-loop weight)
__global__ void deg_kernel(const int* __restrict__ dst, const float* __restrict__ ew,
                           float* __restrict__ deg, int E) {
  int t = blockIdx.x * blockDim.x + threadIdx.x;
  if (t < E)
    __hip_atomic_fetch_add(&deg[dst[t]], ew[t], __ATOMIC_RELAXED, __HIP_MEMORY_SCOPE_AGENT);
}

// in-place deg -> dis = deg>0 ? rsqrt(deg) : 0
__global__ void rsqrt_kernel(float* __restrict__ deg, int N) {
  int i = blockIdx.x * blockDim.x + threadIdx.x;
  if (i < N) {
    float d = deg[i];
    deg[i] = (d > 0.0f) ? rsqrtf(d) : 0.0f;
  }
}

// h[i] = relu(h[i] + b[i % 16])   (HID == 16, row-major)
__global__ void bias_relu16(float* __restrict__ h, const float* __restrict__ b, long n) {
  long i = (long)blockIdx.x * blockDim.x + threadIdx.x;
  if (i < n) {
    float v = h[i] + b[(int)(i & (HID - 1))];
    h[i] = v > 0.0f ? v : 0.0f;
  }
}

// ---------------- GEMM 1: [N,512] x [512,16] via WMMA f32 16x16x4 ----------------
// One wave32 handles a 16-row x 16-col tile; K looped in steps of 4.
// A layout (16x4 f32): lane L -> M = L&15, holds K = k0 + 2*(L>>4) + {0,1}
// B layout (4x16 f32): lane L -> N = L&15, holds K = k0 + 2*(L>>4) + {0,1}
// C/D (16x16 f32):     vgpr r, lane L -> M = r + 8*(L>>4), N = L&15

__global__ void gemm1_wmma(const float* __restrict__ x, const float* __restrict__ W1,
                           float* __restrict__ h, int nrows) {
  int wave = (int)((blockIdx.x * (long)blockDim.x + threadIdx.x) >> 5);
  int lane = threadIdx.x & 31;
  int r0 = wave * 16;
  if (r0 >= nrows) return;              // wave-uniform: EXEC stays all-1s

  int M  = lane & 15;
  int kh = (lane >> 4) * 2;
  const float* arow = x + (size_t)(r0 + M) * F_IN + kh;

  v8f acc = {};
  for (int k0 = 0; k0 < F_IN; k0 += 4) {
    v2f a = *(const v2f*)(arow + k0);
    v2f b;
    b.x = W1[(k0 + kh) * HID + M];      // N index == lane&15
    b.y = W1[(k0 + kh + 1) * HID + M];
    acc = __builtin_amdgcn_wmma_f32_16x16x4_f32(
        /*neg_a=*/false, a, /*neg_b=*/false, b,
        /*c_mod=*/(short)0, acc, /*reuse_a=*/false, /*reuse_b=*/false);
  }

  int nc = lane & 15;
  int mb = (lane >> 4) * 8;
#pragma unroll
  for (int r = 0; r < 8; ++r)
    h[(size_t)(r0 + mb + r) * HID + nc] = acc[r];
}

// ---------------- GEMM 2: [N,16] x [16,40] via WMMA, 3 N-tiles (cols>=40 masked) ---

__global__ void gemm2_wmma(const float* __restrict__ h1, const float* __restrict__ W2,
                           float* __restrict__ h2, int nrows) {
  int wave = (int)((blockIdx.x * (long)blockDim.x + threadIdx.x) >> 5);
  int lane = threadIdx.x & 31;
  int r0 = wave * 16;
  if (r0 >= nrows) return;

  int M  = lane & 15;
  int kh = (lane >> 4) * 2;
  const float* arow = h1 + (size_t)(r0 + M) * HID + kh;

  v8f acc0 = {}, acc1 = {}, acc2 = {};
#pragma unroll
  for (int k0 = 0; k0 < HID; k0 += 4) {
    v2f a = *(const v2f*)(arow + k0);
    int kr0 = (k0 + kh) * NCLS, kr1 = (k0 + kh + 1) * NCLS;

    v2f b0, b1t, b2t;
    int c0 = M;        // tile 0 cols 0..15 (always valid)
    b0.x = W2[kr0 + c0];            b0.y = W2[kr1 + c0];
    int c1 = 16 + M;   // tile 1 cols 16..31 (always valid)
    b1t.x = W2[kr0 + c1];           b1t.y = W2[kr1 + c1];
    int c2 = 32 + M;   // tile 2 cols 32..47 (mask >= 40)
    b2t.x = (c2 < NCLS) ? W2[kr0 + c2] : 0.0f;
    b2t.y = (c2 < NCLS) ? W2[kr1 + c2] : 0.0f;

    acc0 = __builtin_amdgcn_wmma_f32_16x16x4_f32(false, a, false, b0,  (short)0, acc0, false, false);
    acc1 = __builtin_amdgcn_wmma_f32_16x16x4_f32(false, a, false, b1t, (short)0, acc1, false, false);
    acc2 = __builtin_amdgcn_wmma_f32_16x16x4_f32(false, a, false, b2t, (short)0, acc2, false, false);
  }

  int nc = lane & 15;
  int mb = (lane >> 4) * 8;
#pragma unroll
  for (int r = 0; r < 8; ++r) {
    size_t row = (size_t)(r0 + mb + r) * NCLS;
    h2[row + nc]      = acc0[r];
    h2[row + 16 + nc] = acc1[r];
    if (32 + nc < NCLS) h2[row + 32 + nc] = acc2[r];
  }
}

// ---------------- edge scatter-add (covers real edges + self-loops) ----------------
// t < E : edge t;  t >= E : self-loop for node t-E with weight 1.
// h (6.4/16 MB) and agg live in L2 (192 MB), so gathers + fp32 atomics hit L2.

template <int C>
__global__ void scatter_agg(const int* __restrict__ src, const int* __restrict__ dst,
                            const float* __restrict__ ew, const float* __restrict__ dis,
                            const float* __restrict__ h, float* __restrict__ agg,
                            int E, int N) {
  long t = (long)blockIdx.x * blockDim.x + threadIdx.x;
  if (t >= (long)E + N) return;
  int s, d;
  float w;
  if (t < E) { s = src[t]; d = dst[t]; w = ew[t]; }
  else       { s = d = (int)(t - E);   w = 1.0f; }
  float norm = dis[s] * w * dis[d];
  const float* hs = h + (size_t)s * C;
  float* ad = agg + (size_t)d * C;
#pragma unroll
  for (int c = 0; c < C; ++c)
    __hip_atomic_fetch_add(&ad[c], hs[c] * norm, __ATOMIC_RELAXED, __HIP_MEMORY_SCOPE_AGENT);
}

// ---------------- log_softmax over 40 classes, one wave32 per row ----------------

__global__ void logsoftmax40(const float* __restrict__ agg2, const float* __restrict__ b2,
                             float* __restrict__ out, int nrows) {
  int wave = (int)((blockIdx.x * (long)blockDim.x + threadIdx.x) >> 5);
  int lane = threadIdx.x & 31;
  if (wave >= nrows) return;

  const float* row = agg2 + (size_t)wave * NCLS;
  float a0 = row[lane] + b2[lane];                       // lane < 32 < 40
  bool  hi = (lane + 32) < NCLS;                          // lanes 0..7
  float a1 = hi ? (row[lane + 32] + b2[lane + 32]) : -INFINITY;

  float m = fmaxf(a0, a1);
  for (int off = 16; off > 0; off >>= 1)
    m = fmaxf(m, __shfl_xor(m, off, 32));

  float s = expf(a0 - m) + (hi ? expf(a1 - m) : 0.0f);
  for (int off = 16; off > 0; off >>= 1)
    s += __shfl_xor(s, off, 32);
  float ls = logf(s);

  float* orow = out + (size_t)wave * NCLS;
  orow[lane] = a0 - m - ls;
  if (hi) orow[lane + 32] = a1 - m - ls;
}

// ---------------- launch ----------------

static inline int cdiv(long a, int b) { return (int)((a + b - 1) / b); }

extern "C" void kernel_launch(void* const* d_in, const int* in_sizes, int n_in,
                              void* d_out, int out_size, void* d_ws, size_t ws_size,
                              hipStream_t stream) {
  const float* x  = (const float*)d_in[0];
  const int*   ei = (const int*)  d_in[1];
  const float* ew = (const float*)d_in[2];
  const float* W1 = (const float*)d_in[3];
  const float* b1 = (const float*)d_in[4];
  const float* W2 = (const float*)d_in[5];
  const float* b2 = (const float*)d_in[6];

  const int N = in_sizes[0] / F_IN;   // 100000
  const int E = in_sizes[2];          // 3200000
  const int* src = ei;
  const int* dst = ei + E;

  // workspace layout (floats): dis[N] | h1lin[16N] | agg1/h1[16N] | h2lin[40N] | agg2[40N]
  float* wsf   = (float*)d_ws;
  float* dis   = wsf;
  float* h1lin = wsf + (size_t)N;
  float* agg1  = wsf + (size_t)17 * N;   // becomes h1 after bias+relu (in place)
  float* h2lin = wsf + (size_t)33 * N;
  float* agg2  = wsf + (size_t)73 * N;

  const long ET = (long)E + N;           // edges + self-loops

  // 1) init: deg=1 (self-loop weight), agg buffers = 0
  fill_kernel<<<cdiv(N, 256), 256, 0, stream>>>(dis, 1.0f, N);
  fill_kernel<<<cdiv((long)16 * N, 256), 256, 0, stream>>>(agg1, 0.0f, (long)16 * N);
  fill_kernel<<<cdiv((long)NCLS * N, 256), 256, 0, stream>>>(agg2, 0.0f, (long)NCLS * N);

  // 2) degree + symmetric normalization
  deg_kernel<<<cdiv(E, 256), 256, 0, stream>>>(dst, ew, dis, E);
  rsqrt_kernel<<<cdiv(N, 256), 256, 0, stream>>>(dis, N);

  // 3) layer 1 GEMM (WMMA f32): 16 rows per wave
  {
    long threads = ((long)(N + 15) / 16) * 32;
    gemm1_wmma<<<cdiv(threads, 128), 128, 0, stream>>>(x, W1, h1lin, N);
  }
  // 4) layer 1 edge aggregation + bias + relu
  scatter_agg<HID><<<cdiv(ET, 256), 256, 0, stream>>>(src, dst, ew, dis, h1lin, agg1, E, N);
  bias_relu16<<<cdiv((long)16 * N, 256), 256, 0, stream>>>(agg1, b1, (long)16 * N);

  // 5) layer 2 GEMM (WMMA f32)
  {
    long threads = ((long)(N + 15) / 16) * 32;
    gemm2_wmma<<<cdiv(threads, 128), 128, 0, stream>>>(agg1, W2, h2lin, N);
  }
  // 6) layer 2 edge aggregation
  scatter_agg<NCLS><<<cdiv(ET, 256), 256, 0, stream>>>(src, dst, ew, dis, h2lin, agg2, E, N);

  // 7) bias + log_softmax -> d_out
  {
    long threads = (long)N * 32;
    logsoftmax40<<<cdiv(threads, 256), 256, 0, stream>>>(agg2, b2, (float*)d_out, N);
  }
}